// CTCLayer_25838523253028
// MI455X (gfx1250) — compile-verified
//
#include <hip/hip_runtime.h>
#include <hip/hip_bf16.h>
#include <stdint.h>

// Problem constants (from reference setup_inputs)
#define B_   256
#define T_   512
#define C_   256
#define L_   64
#define S_   129          // 2*L+1 extended states
#define BLANK 255         // C-1
#define NEGV (-1e30f)
#define EPSV (1e-7f)

#define PF   8            // async prefetch depth (power of 2)
#define SPAD 132          // padded row length for LDS lp slots

// Low 32 bits of a generic pointer to __shared__ data are the wave-relative
// LDS byte offset (flat-LDS aperture).
__device__ __forceinline__ uint32_t lds_offset_u32(const void* p) {
  return (uint32_t)(uintptr_t)p;
}

// GLOBAL_LOAD_ASYNC_TO_LDS_B32 (VGLOBAL op 96), GV mode:
//   each lane copies one dword from global memory to LDS; tracked by ASYNCcnt
__device__ __forceinline__ void async_load_b32(uint32_t lds_addr, const void* gaddr) {
  asm volatile("global_load_async_to_lds_b32 %0, %1, off"
               :
               : "v"(lds_addr), "v"((uint64_t)(uintptr_t)gaddr)
               : "memory");
}

// S_WAIT_ASYNCCNT <= n (SOPP op 74); n must be a compile-time constant.
__device__ __forceinline__ void wait_asynccnt(int n) {
  asm volatile("s_wait_asynccnt %0" :: "n"(n) : "memory");
}

// Ensure this wave's LDS stores have completed (visible WGP-wide).
__device__ __forceinline__ void wait_dscnt0() {
  asm volatile("s_wait_dscnt 0x0" ::: "memory");
}

// CDNA5 split workgroup barrier (barrier id -1).
__device__ __forceinline__ void wg_barrier_signal() {
  asm volatile("s_barrier_signal -1" ::: "memory");
}
__device__ __forceinline__ void wg_barrier_wait() {
  asm volatile("s_barrier_wait -1" ::: "memory");
}

__global__ __launch_bounds__(160, 8)
void ctc_forward_kernel(const int* __restrict__ y_true,
                        const float* __restrict__ y_pred,
                        float* __restrict__ out) {
  const int  b      = blockIdx.x;
  const int  s      = threadIdx.x;
  const bool active = (s < S_);

  // Async-prefetch ring of raw y_pred values gathered at the extended labels.
  __shared__ float lp[PF][SPAD];
  __shared__ float lpdummy[SPAD];            // sink for tail-of-pipeline dummy issues
  // Double-buffered alpha with 2 leading NEG pads: state s lives at [2+s].
  __shared__ float A[2][S_ + 3];

  // Extended label class for this state + skip-transition mask.
  int  cls  = BLANK;
  bool skip = false;
  if (active && (s & 1)) {
    cls = y_true[(size_t)b * L_ + (s >> 1)];
    if (s >= 3) {
      const int prev = y_true[(size_t)b * L_ + ((s - 3) >> 1)];
      skip = (cls != prev);
    }
  }

  // Per-thread gather column base: y_pred[b, t, cls] = rowbase[t*C]
  const float* rowbase = y_pred + ((size_t)b * T_) * (size_t)C_ + (size_t)cls;

  if (s < 2) { A[0][s] = NEGV; A[1][s] = NEGV; }   // permanent NEG padding

  if (active) {
    // t = 0 init: only s<2 reachable.
    const float y0 = rowbase[0];
    A[0][2 + s] = (s < 2) ? __logf(y0 + EPSV) : NEGV;

    // Prologue: fill the async pipeline with t = 1..PF (slot(t) = (t-1)&(PF-1)).
#pragma unroll
    for (int k = 1; k <= PF; ++k) {
      async_load_b32(lds_offset_u32(&lp[k - 1][s]), rowbase + (size_t)k * C_);
    }
  }
  wait_dscnt0();          // init alpha stores visible
  wg_barrier_signal();    // open round 0

  for (int t = 1; t < T_; ++t) {
    const int rb = (t - 1) & 1;
    const int wb = t & 1;

    // ---- barrier-independent work: overlap with other waves' arrival ----
    float lpv = 0.0f;
    if (active) {
      // Invariant: PF async loads outstanding per wave -> waiting to PF-1
      // guarantees the oldest (the load for timestep t) has landed in LDS.
      wait_asynccnt(PF - 1);

      const int slot = (t - 1) & (PF - 1);
      const float v  = lp[slot][s];

      // Refill the pipeline (dummy issue in the tail keeps ASYNCcnt math uniform).
      if (t + PF <= T_ - 1)
        async_load_b32(lds_offset_u32(&lp[slot][s]), rowbase + (size_t)(t + PF) * C_);
      else
        async_load_b32(lds_offset_u32(&lpdummy[s]), rowbase + (size_t)(T_ - 1) * C_);

      lpv = __logf(v + EPSV);
    }

    // ---- wait for previous round's alpha stores from all waves ----
    wg_barrier_wait();

    if (active) {
      const float a   = A[rb][2 + s];
      const float a1  = A[rb][1 + s];
      const float a2  = skip ? A[rb][s] : NEGV;
      const float m   = fmaxf(a, fmaxf(a1, a2));
      const float sum = __expf(a - m) + __expf(a1 - m) + __expf(a2 - m);
      A[wb][2 + s] = m + __logf(sum) + lpv;
    }

    wait_dscnt0();        // my alpha store visible before signaling
    wg_barrier_signal();  // open round t
  }

  wg_barrier_wait();      // consume final round: all alphaT stores visible

  if (s == 0) {
    const int   fb = (T_ - 1) & 1;
    const float x  = A[fb][2 + S_ - 1];
    const float y  = A[fb][2 + S_ - 2];
    const float m  = fmaxf(x, y);
    out[b] = -(m + __logf(__expf(x - m) + __expf(y - m)));
  }
}

extern "C" void kernel_launch(void* const* d_in, const int* in_sizes, int n_in,
                              void* d_out, int out_size, void* d_ws, size_t ws_size,
                              hipStream_t stream) {
  const int*   y_true = (const int*)d_in[0];    // [B, L] labels (int)
  const float* y_pred = (const float*)d_in[1];  // [B, T, C] probabilities (f32)
  float*       out    = (float*)d_out;          // [B, 1] loss (f32)

  ctc_forward_kernel<<<B_, 160, 0, stream>>>(y_true, y_pred, out);
}